// HATGNN_12429635355039
// MI455X (gfx1250) — compile-verified
//
#include <hip/hip_runtime.h>

typedef float v2f __attribute__((ext_vector_type(2)));
typedef float v8f __attribute__((ext_vector_type(8)));
typedef unsigned int u32x4 __attribute__((ext_vector_type(4)));
typedef unsigned int u32x8 __attribute__((ext_vector_type(8)));

#define CDIM 256
#define KMAX 9
#define USE_ASYNC 1
#define USE_TDM 1

// ---------------------------------------------------------------------------
// Squared row norms: one block (256 threads) per row.
// ---------------------------------------------------------------------------
__global__ void row_norms_kernel(const float* __restrict__ X, float* __restrict__ out) {
    int r = blockIdx.x, c = threadIdx.x;
    __shared__ float red[CDIM];
    float v = X[(size_t)r * CDIM + c];
    red[c] = v * v;
    __syncthreads();
    for (int off = 128; off > 0; off >>= 1) {
        if (c < off) red[c] += red[c + off];
        __syncthreads();
    }
    if (c == 0) out[r] = red[0];
}

// ---------------------------------------------------------------------------
// Async stage of one 16x256 f32 chunk (16 KB) into LDS: 32 lanes x 32
// global_load_async_to_lds_b128 (16 B per lane per op), tracked by ASYNCcnt.
// ---------------------------------------------------------------------------
__device__ __forceinline__ void async_stage_load(const float* gsrc, unsigned lds_off, int lane) {
#if USE_ASYNC
    unsigned long long ga = (unsigned long long)(uintptr_t)gsrc + (unsigned)lane * 16u;
    unsigned la = lds_off + (unsigned)lane * 16u;
    #pragma unroll
    for (int j = 0; j < 32; ++j) {
        asm volatile("global_load_async_to_lds_b128 %0, %1, off"
                     :: "v"(la + (unsigned)(j * 512)),
                        "v"(ga + (unsigned long long)(j * 512))
                     : "memory");
    }
#else
    (void)gsrc; (void)lds_off; (void)lane;
#endif
}

__device__ __forceinline__ void async_wait0() {
#if USE_ASYNC
    asm volatile("s_wait_asynccnt 0x0" ::: "memory");
#endif
}

// ---------------------------------------------------------------------------
// Partial top-9 for a tile of 16 labels over a slice of the candidate stream.
// One wave per block (EXEC all-ones for WMMA).  D = A(cand 16xK) x B(label^T):
// per-lane v2f loads match the f32 A/B striping (lane l -> row l&15, channels
// k0 + 2*(l>>4)).  Score = |p|^2 - 2*dot (|label|^2 drops out of ordering).
// blockIdx.x = label tile, blockIdx.y = candidate slice.
// ---------------------------------------------------------------------------
__global__ __launch_bounds__(32) void topk_kernel(const float* __restrict__ labels,
                                                  const float* __restrict__ cands,
                                                  const float* __restrict__ cnorm,
                                                  int n_cand,
                                                  float* __restrict__ dpart,
                                                  int* __restrict__ ipart,
                                                  int nslices) {
    __shared__ float lbl[16 * CDIM];
    __shared__ float stage[2][16 * CDIM];
    __shared__ float chunk_norm[16];
    __shared__ float md[32 * KMAX];
    __shared__ int   mi[32 * KMAX];

    const int lane = threadIdx.x;
    const int l2 = lane & 15;
    const int hi = lane >> 4;

    // ---- label tile -> LDS via Tensor Data Mover (16x256 f32 tile) ----
#if USE_TDM
    {
        unsigned long long ga =
            (unsigned long long)(uintptr_t)(labels + (size_t)blockIdx.x * 16 * CDIM);
        unsigned lds_lbl = (unsigned)(uintptr_t)&lbl[0];
        // D# group0: count=1 | lds_addr | global_addr[56:0] | type=2
        u32x4 g0 = { 1u, lds_lbl, (unsigned)ga,
                     (unsigned)((ga >> 32) & 0x01FFFFFFu) | 0x80000000u };
        // D# group1: data_size=4B, tensor_dim0=256, tensor_dim1=16,
        //            tile_dim0=256, tile_dim1=16, tensor_dim0_stride=256
        u32x8 g1 = { 0x00020000u,        // mask=0 | data_size=2<<16
                     (256u & 0xFFFFu) << 16,   // tensor_dim0[15:0]
                     (16u << 16),        // tensor_dim0[31:16]=0 | tensor_dim1[15:0]=16
                     (256u << 16),       // tensor_dim1[31:16]=0 | tile_dim0=256
                     16u,                // tile_dim1=16 | tile_dim2=0
                     256u,               // tensor_dim0_stride[31:0]
                     0u, 0u };
        asm volatile("tensor_load_to_lds %0, %1" :: "s"(g0), "s"(g1) : "memory");
        __builtin_amdgcn_s_wait_tensorcnt(0);
    }
#else
    {
        const float4* src = (const float4*)(labels + (size_t)blockIdx.x * 16 * CDIM);
        float4* dst = (float4*)lbl;
        #pragma unroll 4
        for (int j = 0; j < 32; ++j) dst[lane + j * 32] = src[lane + j * 32];
    }
#endif

    float best_d[KMAX];
    int   best_i[KMAX];
    #pragma unroll
    for (int j = 0; j < KMAX; ++j) { best_d[j] = 3.4e38f; best_i[j] = 0x7FFFFFFF; }

    const int nchunks = n_cand >> 4;
    const int c0 = (int)(((long long)blockIdx.y * nchunks) / nslices);
    const int c1 = (int)(((long long)(blockIdx.y + 1) * nchunks) / nslices);

    const unsigned stage_off[2] = { (unsigned)(uintptr_t)&stage[0][0],
                                    (unsigned)(uintptr_t)&stage[1][0] };
#if USE_ASYNC
    if (c0 < c1) async_stage_load(cands + (size_t)c0 * 16 * CDIM, stage_off[0], lane);
#endif

    for (int ch = c0; ch < c1; ++ch) {
        const int cur = (ch - c0) & 1;
        __syncthreads();
        if (lane < 16) chunk_norm[lane] = cnorm[ch * 16 + lane];
#if USE_ASYNC
        async_wait0();  // stage[cur] resident
        if (ch + 1 < c1)
            async_stage_load(cands + (size_t)(ch + 1) * 16 * CDIM, stage_off[cur ^ 1], lane);
#else
        {
            const float4* src = (const float4*)(cands + (size_t)ch * 16 * CDIM);
            float4* dst = (float4*)&stage[cur][0];
            #pragma unroll 4
            for (int j = 0; j < 32; ++j) dst[lane + j * 32] = src[lane + j * 32];
        }
#endif
        __syncthreads();

        // 16x16 dot-product tile via chained f32 WMMA over K=256.
        v8f acc = {0.f, 0.f, 0.f, 0.f, 0.f, 0.f, 0.f, 0.f};
        const float* arow = &stage[cur][l2 * CDIM + 2 * hi];
        const float* brow = &lbl[l2 * CDIM + 2 * hi];
        #pragma unroll 8
        for (int k0 = 0; k0 < CDIM; k0 += 4) {
            v2f a = *(const v2f*)(arow + k0);
            v2f b = *(const v2f*)(brow + k0);
            acc = __builtin_amdgcn_wmma_f32_16x16x4_f32(
                false, a, false, b, (short)0, acc, false, false);
        }

        // D layout: VGPR r -> M = r + 8*hi (candidate), N = lane&15 (label).
        const int base = ch * 16;
        #pragma unroll
        for (int r = 0; r < 8; ++r) {
            const int m = r + 8 * hi;
            const float d = chunk_norm[m] - 2.0f * acc[r];
            const int idv = base + m;
            if (d < best_d[KMAX - 1]) {
                int j = KMAX - 1;
                while (j > 0 && d < best_d[j - 1]) {
                    best_d[j] = best_d[j - 1];
                    best_i[j] = best_i[j - 1];
                    --j;
                }
                best_d[j] = d;
                best_i[j] = idv;
            }
        }
    }

    // Merge lane n (M=0..7 per chunk) with lane n+16 (M=8..15); ties -> lower idx.
    __syncthreads();
    #pragma unroll
    for (int j = 0; j < KMAX; ++j) {
        md[lane * KMAX + j] = best_d[j];
        mi[lane * KMAX + j] = best_i[j];
    }
    __syncthreads();
    if (lane < 16) {
        int pa = lane * KMAX, pb = (lane + 16) * KMAX;
        int ia = 0, ib = 0;
        const size_t s = (size_t)blockIdx.x * 16 + lane;
        const size_t o = (s * nslices + blockIdx.y) * KMAX;
        for (int j = 0; j < KMAX; ++j) {
            float da = md[pa + ia], db = md[pb + ib];
            int xa = mi[pa + ia], xb = mi[pb + ib];
            bool takeA = (da < db) || (da == db && xa < xb);
            dpart[o + j] = takeA ? da : db;
            ipart[o + j] = takeA ? xa : xb;
            if (takeA) ++ia; else ++ib;
        }
    }
}

// ---------------------------------------------------------------------------
// Merge per-slice partial top-9 lists into the global top-9 index list.
// ---------------------------------------------------------------------------
__global__ void merge_topk_kernel(const float* __restrict__ dpart,
                                  const int* __restrict__ ipart,
                                  int nslices, int S, int* __restrict__ tk) {
    int s = blockIdx.x * blockDim.x + threadIdx.x;
    if (s >= S) return;
    float bd[KMAX];
    int   bi[KMAX];
    #pragma unroll
    for (int j = 0; j < KMAX; ++j) { bd[j] = 3.4e38f; bi[j] = 0x7FFFFFFF; }
    for (int sl = 0; sl < nslices; ++sl) {
        size_t base = ((size_t)s * nslices + sl) * KMAX;
        for (int e = 0; e < KMAX; ++e) {
            float d = dpart[base + e];
            int   i = ipart[base + e];
            if (d < bd[KMAX - 1] || (d == bd[KMAX - 1] && i < bi[KMAX - 1])) {
                int j = KMAX - 1;
                while (j > 0 && (d < bd[j - 1] || (d == bd[j - 1] && i < bi[j - 1]))) {
                    bd[j] = bd[j - 1]; bi[j] = bi[j - 1]; --j;
                }
                bd[j] = d; bi[j] = i;
            }
        }
    }
    #pragma unroll
    for (int j = 0; j < KMAX; ++j) tk[(size_t)s * KMAX + j] = bi[j];
}

// ---------------------------------------------------------------------------
// ctx[s] = max_j(cand[idx_j]) - label[s].
// ---------------------------------------------------------------------------
__global__ void gather_ctx_kernel(const float* __restrict__ labels,
                                  const float* __restrict__ cands,
                                  const int* __restrict__ topk,
                                  float* __restrict__ ctx, int k) {
    int s = blockIdx.x, c = threadIdx.x;
    float m = -3.4e38f;
    for (int j = 0; j < k; ++j) {
        int id = topk[(size_t)s * KMAX + j];
        m = fmaxf(m, cands[(size_t)id * CDIM + c]);
    }
    ctx[(size_t)s * CDIM + c] = m - labels[(size_t)s * CDIM + c];
}

// ---------------------------------------------------------------------------
// out = LN(label + concat(label, ctx...) @ W + b).  Block per label row.
// ---------------------------------------------------------------------------
__global__ void linear_ln_kernel(const float* __restrict__ label,
                                 const float* __restrict__ c1,
                                 const float* __restrict__ c2,
                                 const float* __restrict__ c3,
                                 int nctx,
                                 const float* __restrict__ W,
                                 const float* __restrict__ bias,
                                 const float* __restrict__ g,
                                 const float* __restrict__ beta,
                                 float* __restrict__ out) {
    int s = blockIdx.x, c = threadIdx.x;
    __shared__ float in[4 * CDIM];
    __shared__ float red[CDIM];

    float lab = label[(size_t)s * CDIM + c];
    in[c] = lab;
    in[CDIM + c] = c1[(size_t)s * CDIM + c];
    if (nctx >= 2) in[2 * CDIM + c] = c2[(size_t)s * CDIM + c];
    if (nctx >= 3) in[3 * CDIM + c] = c3[(size_t)s * CDIM + c];
    __syncthreads();

    const int D = CDIM * (1 + nctx);
    float acc = lab + bias[c];
    for (int d = 0; d < D; ++d)
        acc = fmaf(in[d], W[(size_t)d * CDIM + c], acc);

    red[c] = acc;
    __syncthreads();
    for (int off = 128; off > 0; off >>= 1) {
        if (c < off) red[c] += red[c + off];
        __syncthreads();
    }
    float mu = red[0] * (1.0f / CDIM);
    __syncthreads();
    float dx = acc - mu;
    red[c] = dx * dx;
    __syncthreads();
    for (int off = 128; off > 0; off >>= 1) {
        if (c < off) red[c] += red[c + off];
        __syncthreads();
    }
    float var = red[0] * (1.0f / CDIM);
    out[(size_t)s * CDIM + c] = dx * rsqrtf(var + 1e-5f) * g[c] + beta[c];
}

// ---------------------------------------------------------------------------
extern "C" void kernel_launch(void* const* d_in, const int* in_sizes, int n_in,
                              void* d_out, int out_size, void* d_ws, size_t ws_size,
                              hipStream_t stream) {
    const float* patch = (const float*)d_in[0];
    const float* mood  = (const float*)d_in[1];
    const float* genre = (const float*)d_in[2];
    const float* sub   = (const float*)d_in[3];
    const float* Wm_w = (const float*)d_in[4];  const float* Wm_b = (const float*)d_in[5];
    const float* Wg_w = (const float*)d_in[6];  const float* Wg_b = (const float*)d_in[7];
    const float* Ws_w = (const float*)d_in[8];  const float* Ws_b = (const float*)d_in[9];
    const float* lnm_g = (const float*)d_in[10]; const float* lnm_b = (const float*)d_in[11];
    const float* lng_g = (const float*)d_in[12]; const float* lng_b = (const float*)d_in[13];
    const float* lns_g = (const float*)d_in[14]; const float* lns_b = (const float*)d_in[15];

    const int NP = in_sizes[0] / CDIM;  // 100000
    float* out = (float*)d_out;
    float* mood_out  = out;
    float* genre_out = out + (size_t)64 * CDIM;
    float* sub_out   = out + (size_t)(64 + 256) * CDIM;

    // Workspace carve-up (fixed part).
    float* ws = (float*)d_ws;
    float* patch_norm = ws;                          // NP
    float* cand_norm  = patch_norm + NP;             // 512
    float* ctx1 = cand_norm + 512;                   // 512*256
    float* ctx2 = ctx1 + 512 * CDIM;
    float* ctx3 = ctx2 + 512 * CDIM;
    int*   tk   = (int*)(ctx3 + 512 * CDIM);         // 512*KMAX
    float* dpart = (float*)(tk + 512 * KMAX);        // 512*SPLIT*KMAX

    // Choose the patch-stream split so the partial lists fit in d_ws.
    size_t fixed_bytes = (size_t)(NP + 512 + 3 * 512 * CDIM) * 4 + (size_t)512 * KMAX * 4;
    int SPLIT = 64;
    while (SPLIT > 1 &&
           fixed_bytes + (size_t)2 * 512 * SPLIT * KMAX * 4 > ws_size)
        SPLIT >>= 1;
    int* ipart = (int*)(dpart + (size_t)512 * SPLIT * KMAX);

    row_norms_kernel<<<NP, CDIM, 0, stream>>>(patch, patch_norm);

    // ---- mood level ----
    topk_kernel<<<dim3(4, SPLIT), 32, 0, stream>>>(mood, patch, patch_norm, NP,
                                                   dpart, ipart, SPLIT);
    merge_topk_kernel<<<1, 64, 0, stream>>>(dpart, ipart, SPLIT, 64, tk);
    gather_ctx_kernel<<<64, CDIM, 0, stream>>>(mood, patch, tk, ctx1, 9);
    linear_ln_kernel<<<64, CDIM, 0, stream>>>(mood, ctx1, nullptr, nullptr, 1,
                                              Wm_w, Wm_b, lnm_g, lnm_b, mood_out);

    // ---- genre level ----
    topk_kernel<<<dim3(16, SPLIT), 32, 0, stream>>>(genre, patch, patch_norm, NP,
                                                    dpart, ipart, SPLIT);
    merge_topk_kernel<<<4, 64, 0, stream>>>(dpart, ipart, SPLIT, 256, tk);
    gather_ctx_kernel<<<256, CDIM, 0, stream>>>(genre, patch, tk, ctx1, 9);
    row_norms_kernel<<<64, CDIM, 0, stream>>>(mood_out, cand_norm);
    topk_kernel<<<dim3(16, 1), 32, 0, stream>>>(genre, mood_out, cand_norm, 64,
                                                dpart, ipart, 1);
    merge_topk_kernel<<<4, 64, 0, stream>>>(dpart, ipart, 1, 256, tk);
    gather_ctx_kernel<<<256, CDIM, 0, stream>>>(genre, mood_out, tk, ctx2, 4);
    linear_ln_kernel<<<256, CDIM, 0, stream>>>(genre, ctx1, ctx2, nullptr, 2,
                                               Wg_w, Wg_b, lng_g, lng_b, genre_out);

    // ---- sub-genre level ----
    topk_kernel<<<dim3(32, SPLIT), 32, 0, stream>>>(sub, patch, patch_norm, NP,
                                                    dpart, ipart, SPLIT);
    merge_topk_kernel<<<8, 64, 0, stream>>>(dpart, ipart, SPLIT, 512, tk);
    gather_ctx_kernel<<<512, CDIM, 0, stream>>>(sub, patch, tk, ctx1, 9);
    row_norms_kernel<<<64, CDIM, 0, stream>>>(mood_out, cand_norm);
    topk_kernel<<<dim3(32, 1), 32, 0, stream>>>(sub, mood_out, cand_norm, 64,
                                                dpart, ipart, 1);
    merge_topk_kernel<<<8, 64, 0, stream>>>(dpart, ipart, 1, 512, tk);
    gather_ctx_kernel<<<512, CDIM, 0, stream>>>(sub, mood_out, tk, ctx2, 3);
    row_norms_kernel<<<256, CDIM, 0, stream>>>(genre_out, cand_norm);
    topk_kernel<<<dim3(32, 1), 32, 0, stream>>>(sub, genre_out, cand_norm, 256,
                                                dpart, ipart, 1);
    merge_topk_kernel<<<8, 64, 0, stream>>>(dpart, ipart, 1, 512, tk);
    gather_ctx_kernel<<<512, CDIM, 0, stream>>>(sub, genre_out, tk, ctx3, 4);
    linear_ln_kernel<<<512, CDIM, 0, stream>>>(sub, ctx1, ctx2, ctx3, 3,
                                               Ws_w, Ws_b, lns_g, lns_b, sub_out);
}